// TargetAwareAttention_13915694039222
// MI455X (gfx1250) — compile-verified
//
#include <hip/hip_runtime.h>

// ---------------------------------------------------------------------------
// TargetAwareAttention for MI455X (gfx1250, wave32)
//   sp = session @ W1 + b1 + b2          [64,128]   (b2 folded into sp)
//   tp = targets @ W2                    [N,128]
//   scores[b,n] = sum_h tanh(sp[b,h] + tp[n,h]) * W3[h] + b3
// GEMMs via V_WMMA_F32_16X16X4_F32 (full fp32 precision, reference is fp32).
// Elementwise phase via hardware v_tanh_f32.
// ---------------------------------------------------------------------------

typedef float v2f __attribute__((ext_vector_type(2)));
typedef float v8f __attribute__((ext_vector_type(8)));

#define EDIM 128
#define HDIM 128
#define BDIM 64
#define TILE_N 128
#define PAD 132   // row stride in floats: %64 != 0 (banks), %4 == 0 (b128 align)

#if defined(__has_builtin)
#if __has_builtin(__builtin_amdgcn_tanhf)
#define FAST_TANH(x) __builtin_amdgcn_tanhf(x)
#endif
#endif
#ifndef FAST_TANH
#define FAST_TANH(x) tanhf(x)
#endif

__global__ __launch_bounds__(256)
void taa_fused_kernel(const float* __restrict__ session,   // [64,128]
                      const float* __restrict__ targets,   // [N,128]
                      const float* __restrict__ W1,        // [128,128]
                      const float* __restrict__ b1,        // [128]
                      const float* __restrict__ W2,        // [128,128]
                      const float* __restrict__ b2,        // [128]
                      const float* __restrict__ W3,        // [128]
                      const float* __restrict__ b3,        // [1]
                      float* __restrict__ out,             // [64,N]
                      int N)
{
    extern __shared__ float lds[];
    float* lds_tp = lds;                        // [TILE_N][PAD]
    float* lds_sp = lds + TILE_N * PAD;         // [BDIM][PAD]
    float* lds_w3 = lds_sp + BDIM * PAD;        // [HDIM]

    const int tid  = threadIdx.x;
    const int lane = tid & 31;
    const int wave = tid >> 5;          // 0..7
    const int m    = lane & 15;         // row (A) / col (B,C) within 16-tile
    const int kg   = lane >> 4;         // 0 or 1: K-group for 32-bit WMMA frags

    const int n_base = blockIdx.x * TILE_N;

    if (tid < HDIM) lds_w3[tid] = W3[tid];

    // ---------------- Phase A1: tp tile (all 8 waves, 16 rows each) --------
    {
        int row = n_base + wave * 16 + m;
        if (row >= N) row = N - 1;                    // clamp; OOB rows unused
        const float* arow = targets + (long)row * EDIM;

        v8f acc[8] = {};
        for (int ks = 0; ks < EDIM / 4; ++ks) {
            const int k0 = ks * 4 + 2 * kg;
            // A 16x4 f32 frag: lane m<16 holds K={k0,k0+1}; lanes 16..31 K+2
            v2f a;
            a.x = arow[k0];
            a.y = arow[k0 + 1];
#pragma unroll
            for (int ct = 0; ct < 8; ++ct) {
                // B 4x16 f32 frag: col = ct*16+m, rows k0,k0+1 per K-group
                v2f b;
                b.x = W2[(k0    ) * HDIM + ct * 16 + m];
                b.y = W2[(k0 + 1) * HDIM + ct * 16 + m];
                acc[ct] = __builtin_amdgcn_wmma_f32_16x16x4_f32(
                    false, a, false, b, (short)0, acc[ct], false, false);
            }
        }
        // C layout: VGPR v, lanes 0-15 -> M=v; lanes 16-31 -> M=v+8; N=m
#pragma unroll
        for (int ct = 0; ct < 8; ++ct) {
#pragma unroll
            for (int v = 0; v < 8; ++v) {
                int r = wave * 16 + v + 8 * kg;
                lds_tp[r * PAD + ct * 16 + m] = acc[ct][v];
            }
        }
    }

    // ---------------- Phase A2: sp (waves 0..3, 16 rows each) --------------
    if (wave < 4) {
        const float* arow = session + (wave * 16 + m) * EDIM;
        v8f acc[8] = {};
        for (int ks = 0; ks < EDIM / 4; ++ks) {
            const int k0 = ks * 4 + 2 * kg;
            v2f a;
            a.x = arow[k0];
            a.y = arow[k0 + 1];
#pragma unroll
            for (int ct = 0; ct < 8; ++ct) {
                v2f b;
                b.x = W1[(k0    ) * HDIM + ct * 16 + m];
                b.y = W1[(k0 + 1) * HDIM + ct * 16 + m];
                acc[ct] = __builtin_amdgcn_wmma_f32_16x16x4_f32(
                    false, a, false, b, (short)0, acc[ct], false, false);
            }
        }
#pragma unroll
        for (int ct = 0; ct < 8; ++ct) {
            const int col  = ct * 16 + m;
            const float bias = b1[col] + b2[col];   // fold both biases into sp
#pragma unroll
            for (int v = 0; v < 8; ++v) {
                int r = wave * 16 + v + 8 * kg;
                lds_sp[r * PAD + col] = acc[ct][v] + bias;
            }
        }
    }

    __syncthreads();

    // ---------------- Phase B: tanh-reduce against W3 ----------------------
    const float b3v  = b3[0];
    const int   nl   = tid & (TILE_N - 1);   // target column owned by thread
    const int   half = tid >> 7;             // 0 or 1: batch-row interleave
    const int   n    = n_base + nl;
    const bool  valid = (n < N);

    const float4* tpv = (const float4*)(lds_tp + nl * PAD);   // 16B aligned
    const float4* w3v = (const float4*)(lds_w3);

    for (int b = half; b < BDIM; b += 2) {
        const float4* spv = (const float4*)(lds_sp + b * PAD); // broadcast
        float acc = 0.0f;
#pragma unroll 8
        for (int h4 = 0; h4 < HDIM / 4; ++h4) {
            float4 t = tpv[h4];
            float4 s = spv[h4];
            float4 w = w3v[h4];
            acc += FAST_TANH(s.x + t.x) * w.x;
            acc += FAST_TANH(s.y + t.y) * w.y;
            acc += FAST_TANH(s.z + t.z) * w.z;
            acc += FAST_TANH(s.w + t.w) * w.w;
        }
        if (valid) out[(long)b * N + n] = acc + b3v;
    }
}

extern "C" void kernel_launch(void* const* d_in, const int* in_sizes, int n_in,
                              void* d_out, int out_size, void* d_ws, size_t ws_size,
                              hipStream_t stream) {
    const float* session = (const float*)d_in[0];
    const float* targets = (const float*)d_in[1];
    const float* W1      = (const float*)d_in[2];
    const float* b1      = (const float*)d_in[3];
    const float* W2      = (const float*)d_in[4];
    const float* b2      = (const float*)d_in[5];
    const float* W3      = (const float*)d_in[6];
    const float* b3      = (const float*)d_in[7];
    float* out = (float*)d_out;

    const int N = in_sizes[1] / EDIM;                 // 20000
    const int blocks = (N + TILE_N - 1) / TILE_N;     // 157
    const size_t shmem =
        (size_t)(TILE_N * PAD + BDIM * PAD + HDIM) * sizeof(float); // ~99.5 KB

    hipLaunchKernelGGL(taa_fused_kernel, dim3(blocks), dim3(256), shmem, stream,
                       session, targets, W1, b1, W2, b2, W3, b3, out, N);
}